// AtomLearningModule_54228257079794
// MI455X (gfx1250) — compile-verified
//
#include <hip/hip_runtime.h>

typedef __attribute__((ext_vector_type(2))) float v2f;
typedef __attribute__((ext_vector_type(8))) float v8f;

#define HID 128
#define LDW 144   // padded LDS row stride (floats) to avoid bank conflicts

// ---------------------------------------------------------------------------
// GEMM: out[n,128] = in[n,128] @ W[128,128]   (exact fp32 via WMMA 16x16x4)
// block = 128 threads = 4 waves; each block does a 16-row slab, each wave two
// 16x16 column tiles. W staged once into LDS.
// ---------------------------------------------------------------------------
__global__ __launch_bounds__(128) void gcn_gemm(const float* __restrict__ in,
                                                const float* __restrict__ W,
                                                float* __restrict__ out, int n)
{
    __shared__ float lw[HID * LDW];
    const int tid = threadIdx.x;
    // cooperative load of W (128x128 f32) into padded LDS
    for (int idx = tid; idx < HID * (HID / 4); idx += 128) {
        const int row = idx >> 5;           // 32 float4 per row
        const int c4  = (idx & 31) << 2;
        const float4 v = *(const float4*)(W + row * HID + c4);
        float* p = &lw[row * LDW + c4];
        p[0] = v.x; p[1] = v.y; p[2] = v.z; p[3] = v.w;
    }
    __syncthreads();

    const int lane = tid & 31;
    const int wave = tid >> 5;
    const int m    = lane & 15;             // A row in tile / B,D col in tile
    const int kof  = (lane >> 4) << 1;      // 0 for lanes 0-15, 2 for 16-31
    const int cb   = wave * 32;             // two 16-col tiles per wave

    int arow = blockIdx.x * 16 + m;
    if (arow >= n) arow = n - 1;            // clamp (keeps EXEC all-1 for WMMA)
    const float* __restrict__ ap = in + (size_t)arow * HID;

    v8f acc0 = {}; v8f acc1 = {};
#pragma unroll
    for (int k0 = 0; k0 < HID; k0 += 4) {
        v2f a, b0, b1;
        a.x  = ap[k0 + kof + 0];
        a.y  = ap[k0 + kof + 1];
        b0.x = lw[(k0 + kof + 0) * LDW + cb + m];
        b0.y = lw[(k0 + kof + 1) * LDW + cb + m];
        b1.x = lw[(k0 + kof + 0) * LDW + cb + 16 + m];
        b1.y = lw[(k0 + kof + 1) * LDW + cb + 16 + m];
        acc0 = __builtin_amdgcn_wmma_f32_16x16x4_f32(false, a, false, b0,
                                                     (short)0, acc0, false, false);
        acc1 = __builtin_amdgcn_wmma_f32_16x16x4_f32(false, a, false, b1,
                                                     (short)0, acc1, false, false);
    }

    // D layout: VGPR v -> M = v + 8*(lane>=16), N = lane&15
    const int r0 = blockIdx.x * 16 + ((lane >> 4) << 3);
#pragma unroll
    for (int v = 0; v < 8; ++v) {
        const int r = r0 + v;
        if (r < n) {
            out[(size_t)r * HID + cb + m]      = acc0[v];
            out[(size_t)r * HID + cb + 16 + m] = acc1[v];
        }
    }
}

// ---------------------------------------------------------------------------
// degree / normalization
// ---------------------------------------------------------------------------
__global__ void deg_init(float* __restrict__ deg, int n) {
    int i = blockIdx.x * blockDim.x + threadIdx.x;
    if (i < n) deg[i] = 1.0f;               // self-loop contributes 1
}
__global__ void deg_accum(const long long* __restrict__ dst,
                          float* __restrict__ deg, int e) {
    int i = blockIdx.x * blockDim.x + threadIdx.x;
    if (i < e) atomicAdd(&deg[(int)dst[i]], 1.0f);
}
__global__ void deg_rsqrt(float* __restrict__ deg, int n) {
    int i = blockIdx.x * blockDim.x + threadIdx.x;
    if (i < n) deg[i] = rsqrtf(deg[i]);     // in place: deg -> dinv
}

// ---------------------------------------------------------------------------
// agg init with self-loop term: agg[i] = h2[i] * dinv[i]^2   (float4/thread)
// ---------------------------------------------------------------------------
__global__ void selfloop_init(const float* __restrict__ h2,
                              const float* __restrict__ dinv,
                              float* __restrict__ agg, int n) {
    int i = blockIdx.x * blockDim.x + threadIdx.x;     // n*32 threads
    if (i >= n * (HID / 4)) return;
    const int node = i >> 5;
    const int c4   = (i & 31) << 2;
    const float d  = dinv[node];
    const float nr = d * d;
    const float4 v = *(const float4*)(h2 + (size_t)node * HID + c4);
    float4 o; o.x = v.x * nr; o.y = v.y * nr; o.z = v.z * nr; o.w = v.w * nr;
    *(float4*)(agg + (size_t)node * HID + c4) = o;
}

// ---------------------------------------------------------------------------
// edge scatter: one wave32 per edge, lane -> float4 of the 128-wide feature.
// h2/agg are L2-resident (51 MB each < 192 MB L2) -> L2 atomic bound.
// ---------------------------------------------------------------------------
__global__ __launch_bounds__(256) void edge_agg(const long long* __restrict__ src,
                                                const long long* __restrict__ dst,
                                                const float* __restrict__ dinv,
                                                const float* __restrict__ h2,
                                                float* __restrict__ agg, int e) {
    const int edge = blockIdx.x * 8 + (threadIdx.x >> 5);
    if (edge >= e) return;
    const int s = (int)src[edge];
    const int d = (int)dst[edge];
    const float nrm = dinv[s] * dinv[d];
    const int c4 = (threadIdx.x & 31) << 2;
    const float4 v = *(const float4*)(h2 + (size_t)s * HID + c4);
    float* a = agg + (size_t)d * HID + c4;
    atomicAdd(a + 0, v.x * nrm);
    atomicAdd(a + 1, v.y * nrm);
    atomicAdd(a + 2, v.z * nrm);
    atomicAdd(a + 3, v.w * nrm);
}

// ---------------------------------------------------------------------------
// h = relu(agg + bias)
// ---------------------------------------------------------------------------
__global__ void relu_bias(const float* __restrict__ agg,
                          const float* __restrict__ bias,
                          float* __restrict__ h, int n) {
    int i = blockIdx.x * blockDim.x + threadIdx.x;     // n*32 threads
    if (i >= n * (HID / 4)) return;
    const int node = i >> 5;
    const int c4   = (i & 31) << 2;
    const float4 v = *(const float4*)(agg + (size_t)node * HID + c4);
    const float4 b = *(const float4*)(bias + c4);
    float4 o;
    o.x = fmaxf(v.x + b.x, 0.0f);
    o.y = fmaxf(v.y + b.y, 0.0f);
    o.z = fmaxf(v.z + b.z, 0.0f);
    o.w = fmaxf(v.w + b.w, 0.0f);
    *(float4*)(h + (size_t)node * HID + c4) = o;
}

// ---------------------------------------------------------------------------
// global mean pool
// ---------------------------------------------------------------------------
__global__ void pool_zero(float* __restrict__ out, float* __restrict__ cnt,
                          int g) {
    int i = blockIdx.x * blockDim.x + threadIdx.x;
    if (i < g * HID) out[i] = 0.0f;
    else if (i < g * HID + g) cnt[i - g * HID] = 0.0f;
}
__global__ __launch_bounds__(256) void pool_accum(const float* __restrict__ h,
                                                  const long long* __restrict__ batch,
                                                  float* __restrict__ out,
                                                  float* __restrict__ cnt, int n) {
    const int node = blockIdx.x * 8 + (threadIdx.x >> 5);
    if (node >= n) return;
    const int g    = (int)batch[node];
    const int lane = threadIdx.x & 31;
    const int c4   = lane << 2;
    const float4 v = *(const float4*)(h + (size_t)node * HID + c4);
    float* o = out + (size_t)g * HID + c4;
    atomicAdd(o + 0, v.x);
    atomicAdd(o + 1, v.y);
    atomicAdd(o + 2, v.z);
    atomicAdd(o + 3, v.w);
    if (lane == 0) atomicAdd(&cnt[g], 1.0f);
}
__global__ void pool_div(float* __restrict__ out, const float* __restrict__ cnt,
                         int g) {
    int i = blockIdx.x * blockDim.x + threadIdx.x;
    if (i < g * HID) out[i] /= fmaxf(cnt[i >> 7], 1.0f);
}

// ---------------------------------------------------------------------------
extern "C" void kernel_launch(void* const* d_in, const int* in_sizes, int n_in,
                              void* d_out, int out_size, void* d_ws, size_t ws_size,
                              hipStream_t stream) {
    const float*     x  = (const float*)d_in[0];
    const long long* ei = (const long long*)d_in[1];   // int64 [2,E]
    const long long* bt = (const long long*)d_in[2];   // int64 [N]
    const float*     W0 = (const float*)d_in[3];
    const float*     Wh = (const float*)d_in[4];
    const float*     bb = (const float*)d_in[5];

    const int N = in_sizes[0] / HID;
    const int E = in_sizes[1] / 2;
    const int L = in_sizes[5] / HID;
    const int G = out_size / HID;

    const long long* src = ei;
    const long long* dst = ei + E;

    // workspace carve-out
    float* h    = (float*)d_ws;                         // N*128
    float* h2   = h   + (size_t)N * HID;                // N*128
    float* agg  = h2  + (size_t)N * HID;                // N*128
    float* dinv = agg + (size_t)N * HID;                // N (deg -> dinv)
    float* cnt  = dinv + N;                             // G
    float* out  = (float*)d_out;

    const int nodeVecThreads = N * (HID / 4);           // one float4 each
    const int nvBlocks = (nodeVecThreads + 255) / 256;

    // degree + normalization
    deg_init  <<<(N + 255) / 256, 256, 0, stream>>>(dinv, N);
    deg_accum <<<(E + 255) / 256, 256, 0, stream>>>(dst, dinv, E);
    deg_rsqrt <<<(N + 255) / 256, 256, 0, stream>>>(dinv, N);

    // 7 GCN layers
    for (int l = 0; l < L; ++l) {
        const float* hin = (l == 0) ? x : h;
        const float* W   = (l == 0) ? W0 : (Wh + (size_t)(l - 1) * HID * HID);
        const float* bl  = bb + (size_t)l * HID;

        gcn_gemm <<<(N + 15) / 16, 128, 0, stream>>>(hin, W, h2, N);
        selfloop_init <<<nvBlocks, 256, 0, stream>>>(h2, dinv, agg, N);
        edge_agg <<<(E + 7) / 8, 256, 0, stream>>>(src, dst, dinv, h2, agg, E);
        relu_bias <<<nvBlocks, 256, 0, stream>>>(agg, bl, h, N);
    }

    // global mean pool -> d_out
    pool_zero  <<<(G * HID + G + 255) / 256, 256, 0, stream>>>(out, cnt, G);
    pool_accum <<<(N + 7) / 8, 256, 0, stream>>>(h, bt, out, cnt, N);
    pool_div   <<<(G * HID + 255) / 256, 256, 0, stream>>>(out, cnt, G);
}